// GATLayer_51788715655659
// MI455X (gfx1250) — compile-verified
//
#include <hip/hip_runtime.h>
#include <hip/hip_bf16.h>
#include <math.h>

#define N_NODES   50000
#define N_EDGES   1600000
#define IN_DIM    128
#define HEADS     8
#define OUT_C     16
#define NEG_SLOPE 0.2f
#define BN_EPS    1e-5f

typedef __attribute__((ext_vector_type(2))) float v2f;
typedef __attribute__((ext_vector_type(8))) float v8f;

// ---------------------------------------------------------------------------
// Kernel 1: xl = x @ W via V_WMMA_F32_16X16X4_F32.
// grid = (3125 row-tiles, 8 col-tiles == heads), block = 32 (one wave32).
// ---------------------------------------------------------------------------
__global__ void gat_gemm_wmma(const float* __restrict__ x,
                              const float* __restrict__ W,
                              float* __restrict__ xl) {
    const int mt   = blockIdx.x;          // node tile (16 rows)
    const int nt   = blockIdx.y;          // channel tile (16 cols) == head
    const int lane = threadIdx.x;         // 0..31
    const int half = lane >> 4;           // 0: K pair {0,1}; 1: K pair {2,3}
    const int l16  = lane & 15;

    const float* __restrict__ xrow = x + (size_t)(mt * 16 + l16) * IN_DIM;
    const float* __restrict__ wcol = W + nt * 16 + l16;

    v8f acc = {};
    #pragma unroll 4
    for (int kk = 0; kk < IN_DIM; kk += 4) {
        const int k0 = kk + 2 * half;
        v2f a, b;
        a.x = xrow[k0];
        a.y = xrow[k0 + 1];
        b.x = wcol[(size_t)k0 * IN_DIM];
        b.y = wcol[(size_t)(k0 + 1) * IN_DIM];
        acc = __builtin_amdgcn_wmma_f32_16x16x4_f32(
            /*neg_a=*/false, a, /*neg_b=*/false, b,
            /*c_mod=*/(short)0, acc, /*reuse_a=*/false, /*reuse_b=*/false);
    }

    // C/D layout: lane l, VGPR v -> M = 8*(l/16)+v, N = l%16
    float* __restrict__ outp = xl + (size_t)(mt * 16) * IN_DIM + nt * 16 + l16;
    #pragma unroll
    for (int v = 0; v < 8; ++v)
        outp[(size_t)(half * 8 + v) * IN_DIM] = acc[v];
}

// ---------------------------------------------------------------------------
// Kernel 2: per-(node, head) attention logits a_src / a_dst.
// ---------------------------------------------------------------------------
__global__ void gat_att(const float* __restrict__ xl,
                        const float* __restrict__ att_src,
                        const float* __restrict__ att_dst,
                        float* __restrict__ a_src,
                        float* __restrict__ a_dst) {
    int idx = blockIdx.x * blockDim.x + threadIdx.x;   // n*HEADS + h
    if (idx >= N_NODES * HEADS) return;
    const int n = idx >> 3, h = idx & 7;
    const float* __restrict__ v  = xl + (size_t)n * IN_DIM + h * OUT_C;
    const float* __restrict__ as = att_src + h * OUT_C;
    const float* __restrict__ ad = att_dst + h * OUT_C;
    float s = 0.f, d = 0.f;
    #pragma unroll
    for (int c = 0; c < OUT_C; ++c) { float t = v[c]; s += t * as[c]; d += t * ad[c]; }
    a_src[idx] = s;
    a_dst[idx] = d;
}

// ---------------------------------------------------------------------------
// Init: maxbuf = -inf
// ---------------------------------------------------------------------------
__global__ void gat_init_max(float* __restrict__ maxbuf) {
    int idx = blockIdx.x * blockDim.x + threadIdx.x;
    if (idx < N_NODES * HEADS) maxbuf[idx] = -INFINITY;
}

__device__ __forceinline__ float leaky(float v) {
    return v > 0.f ? v : v * NEG_SLOPE;
}

// Float atomic max via sign-split int/uint trick (valid with -inf init).
__device__ __forceinline__ void atomicMaxF(float* addr, float value) {
    if (value >= 0.f)
        atomicMax((int*)addr, __float_as_int(value));
    else
        atomicMin((unsigned int*)addr, __float_as_uint(value));
}

// ---------------------------------------------------------------------------
// Kernel 3a: segment max of leaky_relu(a_src[src]+a_dst[dst]) over dst.
// ---------------------------------------------------------------------------
__global__ void gat_edge_max(const int* __restrict__ src,
                             const int* __restrict__ dst,
                             const float* __restrict__ a_src,
                             const float* __restrict__ a_dst,
                             float* __restrict__ maxbuf) {
    int idx = blockIdx.x * blockDim.x + threadIdx.x;   // e*HEADS + h
    if (idx >= N_EDGES * HEADS) return;
    const int e = idx >> 3, h = idx & 7;
    const int s = src[e], d = dst[e];
    const float v = leaky(a_src[s * HEADS + h] + a_dst[d * HEADS + h]);
    atomicMaxF(&maxbuf[d * HEADS + h], v);
}

// ---------------------------------------------------------------------------
// Kernel 3b: denom[dst] += exp(e - max[dst])
// ---------------------------------------------------------------------------
__global__ void gat_edge_expsum(const int* __restrict__ src,
                                const int* __restrict__ dst,
                                const float* __restrict__ a_src,
                                const float* __restrict__ a_dst,
                                const float* __restrict__ maxbuf,
                                float* __restrict__ denom) {
    int idx = blockIdx.x * blockDim.x + threadIdx.x;
    if (idx >= N_EDGES * HEADS) return;
    const int e = idx >> 3, h = idx & 7;
    const int s = src[e], d = dst[e];
    const float v = leaky(a_src[s * HEADS + h] + a_dst[d * HEADS + h]);
    atomicAdd(&denom[d * HEADS + h], expf(v - maxbuf[d * HEADS + h]));
}

// ---------------------------------------------------------------------------
// Kernel 3c: out[dst] += alpha * xl[src]   (16 channels per (edge, head))
// ---------------------------------------------------------------------------
__global__ void gat_edge_scatter(const int* __restrict__ src,
                                 const int* __restrict__ dst,
                                 const float* __restrict__ a_src,
                                 const float* __restrict__ a_dst,
                                 const float* __restrict__ maxbuf,
                                 const float* __restrict__ denom,
                                 const float* __restrict__ xl,
                                 float* __restrict__ outacc) {
    int idx = blockIdx.x * blockDim.x + threadIdx.x;
    if (idx >= N_EDGES * HEADS) return;
    const int e = idx >> 3, h = idx & 7;
    const int s = src[e], d = dst[e];
    const float v = leaky(a_src[s * HEADS + h] + a_dst[d * HEADS + h]);
    const float alpha = expf(v - maxbuf[d * HEADS + h]) /
                        (denom[d * HEADS + h] + 1e-16f);
    const float* __restrict__ xs = xl + (size_t)s * IN_DIM + h * OUT_C;
    float* __restrict__ od = outacc + (size_t)d * IN_DIM + h * OUT_C;
    #pragma unroll
    for (int c = 0; c < OUT_C; ++c)
        atomicAdd(&od[c], alpha * xs[c]);
}

// ---------------------------------------------------------------------------
// Kernel 4: per-channel sum & sumsq of (outacc + bias).
// block = 128 threads (one per channel), each block covers a slab of nodes.
// ---------------------------------------------------------------------------
#define STAT_SLAB 256
__global__ void gat_bn_stats(const float* __restrict__ outacc,
                             const float* __restrict__ bias,
                             float* __restrict__ sums,
                             float* __restrict__ sumsq) {
    const int c = threadIdx.x;                 // 0..127
    const int n0 = blockIdx.x * STAT_SLAB;
    const int n1 = min(n0 + STAT_SLAB, N_NODES);
    const float b = bias[c];
    float s = 0.f, sq = 0.f;
    for (int n = n0; n < n1; ++n) {
        const float v = outacc[(size_t)n * IN_DIM + c] + b;
        s += v; sq += v * v;
    }
    atomicAdd(&sums[c], s);
    atomicAdd(&sumsq[c], sq);
}

// ---------------------------------------------------------------------------
// Kernel 5: normalize + relu + residual -> d_out
// ---------------------------------------------------------------------------
__global__ void gat_finalize(const float* __restrict__ x,
                             const float* __restrict__ outacc,
                             const float* __restrict__ bias,
                             const float* __restrict__ gamma,
                             const float* __restrict__ beta,
                             const float* __restrict__ sums,
                             const float* __restrict__ sumsq,
                             float* __restrict__ out) {
    int idx = blockIdx.x * blockDim.x + threadIdx.x;
    if (idx >= N_NODES * IN_DIM) return;
    const int c = idx & (IN_DIM - 1);
    const float invN = 1.0f / (float)N_NODES;
    const float mean = sums[c] * invN;
    const float var  = sumsq[c] * invN - mean * mean;
    const float h    = outacc[idx] + bias[c];
    float hn = (h - mean) * rsqrtf(var + BN_EPS) * gamma[c] + beta[c];
    hn = fmaxf(hn, 0.f);
    out[idx] = x[idx] + hn;
}

// ---------------------------------------------------------------------------
// Host-side launch
// ---------------------------------------------------------------------------
extern "C" void kernel_launch(void* const* d_in, const int* in_sizes, int n_in,
                              void* d_out, int out_size, void* d_ws, size_t ws_size,
                              hipStream_t stream) {
    const float* x        = (const float*)d_in[0];
    const int*   ei       = (const int*)d_in[1];      // [2, E]
    const float* W        = (const float*)d_in[2];
    const float* att_src  = (const float*)d_in[3];
    const float* att_dst  = (const float*)d_in[4];
    const float* bias     = (const float*)d_in[5];
    const float* gamma    = (const float*)d_in[6];
    const float* beta     = (const float*)d_in[7];
    float* out = (float*)d_out;

    const int* src = ei;
    const int* dst = ei + N_EDGES;

    // Workspace layout (floats)
    char* ws = (char*)d_ws;
    float* xl      = (float*)ws;                               // N*128
    float* a_src_b = xl + (size_t)N_NODES * IN_DIM;            // N*8
    float* a_dst_b = a_src_b + (size_t)N_NODES * HEADS;        // N*8
    float* maxbuf  = a_dst_b + (size_t)N_NODES * HEADS;        // N*8
    float* denom   = maxbuf + (size_t)N_NODES * HEADS;         // N*8
    float* outacc  = denom + (size_t)N_NODES * HEADS;          // N*128
    float* sums    = outacc + (size_t)N_NODES * IN_DIM;        // 128
    float* sumsq   = sums + IN_DIM;                            // 128

    // Zero accumulators (graph-capture legal)
    hipMemsetAsync(denom, 0, (size_t)N_NODES * HEADS * sizeof(float), stream);
    hipMemsetAsync(outacc, 0, (size_t)N_NODES * IN_DIM * sizeof(float), stream);
    hipMemsetAsync(sums, 0, 2 * IN_DIM * sizeof(float), stream);

    // 1) GEMM via WMMA: 3125 row tiles x 8 head tiles, one wave each
    gat_gemm_wmma<<<dim3(N_NODES / 16, HEADS), 32, 0, stream>>>(x, W, xl);

    // 2) attention logits
    {
        int total = N_NODES * HEADS;
        gat_att<<<(total + 255) / 256, 256, 0, stream>>>(xl, att_src, att_dst,
                                                         a_src_b, a_dst_b);
        gat_init_max<<<(total + 255) / 256, 256, 0, stream>>>(maxbuf);
    }

    // 3) edge phase
    {
        int total = N_EDGES * HEADS;
        int blocks = (total + 255) / 256;
        gat_edge_max<<<blocks, 256, 0, stream>>>(src, dst, a_src_b, a_dst_b, maxbuf);
        gat_edge_expsum<<<blocks, 256, 0, stream>>>(src, dst, a_src_b, a_dst_b,
                                                    maxbuf, denom);
        gat_edge_scatter<<<blocks, 256, 0, stream>>>(src, dst, a_src_b, a_dst_b,
                                                     maxbuf, denom, xl, outacc);
    }

    // 4) BN statistics
    gat_bn_stats<<<(N_NODES + STAT_SLAB - 1) / STAT_SLAB, IN_DIM, 0, stream>>>(
        outacc, bias, sums, sumsq);

    // 5) normalize + relu + residual
    {
        int total = N_NODES * IN_DIM;
        gat_finalize<<<(total + 255) / 256, 256, 0, stream>>>(
            x, outacc, bias, gamma, beta, sums, sumsq, out);
    }
}